// TreecrfLossORL_44659069943949
// MI455X (gfx1250) — compile-verified
//
#include <hip/hip_runtime.h>
#include <cstdint>
#include <math.h>

#define NEGV (-1000000000.0f)
constexpr int Bn = 16;
constexpr int Nn = 256;
constexpr int PITCH = Nn + 1; // 257: column stride 257 mod 64 banks = 1 -> conflict-free

// logaddexp matching jnp.logaddexp semantics for finite inputs
__device__ __forceinline__ float lae(float a, float b) {
  float M = fmaxf(a, b);
  float mn = fminf(a, b);
  return M + log1pf(__expf(mn - M));
}

// ---- CDNA5 async global -> LDS staging (ASYNCcnt path, CDNA5 ISA) ----
__device__ __forceinline__ void async_ld_b64(uint32_t lds_off, const void* gaddr) {
  asm volatile("global_load_async_to_lds_b64 %0, %1, off"
               :: "v"(lds_off), "v"((uint64_t)(uintptr_t)gaddr) : "memory");
}
__device__ __forceinline__ void async_ld_b8(uint32_t lds_off, const void* gaddr) {
  asm volatile("global_load_async_to_lds_b8 %0, %1, off"
               :: "v"(lds_off), "v"((uint64_t)(uintptr_t)gaddr) : "memory");
}
__device__ __forceinline__ void wait_async0() {
  asm volatile("s_wait_asynccnt 0x0" ::: "memory");
}
// generic (flat) shared pointer -> LDS byte offset: low 32 bits of flat LDS-aperture addr
__device__ __forceinline__ uint32_t lds_off(const void* p) {
  return (uint32_t)(uintptr_t)p;
}

__global__ __launch_bounds__(512)
void cyk_kernel(const float* __restrict__ logits,        // [B,N,N,2]
                const unsigned char* __restrict__ sind,  // spans_ind  [B,N,N]
                const unsigned char* __restrict__ mspan, // maskspan   [B,N,N]
                const unsigned char* __restrict__ smask, // span_mask  [B,N,N]
                float* __restrict__ res) {               // [2*B] chart roots
  extern __shared__ char smem[];
  float* alpha = (float*)smem;                                   // PITCH*Nn floats (263168 B)
  float* slog  = (float*)(smem + (size_t)PITCH * Nn * 4);        // 2 bufs * Nn * 2 floats
  unsigned char* sbi = (unsigned char*)(smem + (size_t)PITCH*Nn*4 + 2*Nn*2*4); // 2*Nn
  unsigned char* sbm = sbi + 2 * Nn;                             // 2*Nn

  const int b    = blockIdx.x;
  const int pass = blockIdx.y;        // 0: ob (masked), 1: allp
  const int tid  = threadIdx.x;
  const size_t base = (size_t)b * Nn * Nn;

  // ---- init chart: diagonal = s[i,i], rest = NEG ----
  for (int idx = tid; idx < Nn * Nn; idx += blockDim.x) {
    int i = idx >> 8, j = idx & (Nn - 1);
    float v = NEGV;
    if (i == j) {
      size_t o = base + (size_t)i * Nn + j;
      float l0 = logits[2 * o], l1 = logits[2 * o + 1];
      if (pass) {
        v = lae(l0, l1);
      } else {
        bool sm = smask[o] != 0, si = sind[o] != 0;
        float c0 = (sm || si)  ? NEGV : l0;
        float c1 = (sm || !si) ? NEGV : l1;
        v = lae(c0, c1);
      }
    }
    alpha[i * PITCH + j] = v;
  }

  // ---- prestage score diagonal for width 2 into buffer 0 ----
  if (tid < Nn - 1) {
    int i = tid, j = i + 1;
    size_t o = base + (size_t)i * Nn + j;
    async_ld_b64(lds_off(&slog[2 * i]),      &logits[2 * o]);
    async_ld_b8 (lds_off(&sbi[i]),           &sind[o]);
    async_ld_b8 (lds_off(&sbm[i]),           &smask[o]);
  }
  __syncthreads();

  const int wave = tid >> 5, lane = tid & 31, nwaves = blockDim.x >> 5;

  for (int w = 2; w <= Nn; ++w) {
    wait_async0();        // drain this wave's staged loads for width w
    __syncthreads();      // all waves' staged data now visible in LDS
    const int cur = w & 1;

    // issue async loads for width w+1 into the other buffer (overlaps compute below)
    if (w < Nn && tid < Nn - w) {
      int i = tid, j = i + w;
      int nxt = cur ^ 1;
      size_t o = base + (size_t)i * Nn + j;
      async_ld_b64(lds_off(&slog[nxt * 2 * Nn + 2 * i]), &logits[2 * o]);
      async_ld_b8 (lds_off(&sbi[nxt * Nn + i]),          &sind[o]);
      async_ld_b8 (lds_off(&sbm[nxt * Nn + i]),          &smask[o]);
    }

    // one wave per span; lanes cover split points with online logsumexp
    for (int i = wave; i <= Nn - w; i += nwaves) {
      const int j = i + w - 1;
      float m = -3.0e38f, acc = 0.0f;
      const float* rowL = &alpha[i * PITCH + i];
      for (int u = lane; u <= w - 2; u += 32) {
        float t = rowL[u] + alpha[(i + u + 1) * PITCH + j];
        if (t > m) { acc = acc * __expf(m - t) + 1.0f; m = t; }
        else       { acc += __expf(t - m); }
      }
      // wave32 butterfly merge of (max, scaled-sum)
      for (int off = 16; off; off >>= 1) {
        float m2 = __shfl_xor(m, off, 32);
        float a2 = __shfl_xor(acc, off, 32);
        float M  = fmaxf(m, m2);
        acc = acc * __expf(m - M) + a2 * __expf(m2 - M);
        m = M;
      }
      if (lane == 0) {
        float l0 = slog[cur * 2 * Nn + 2 * i];
        float l1 = slog[cur * 2 * Nn + 2 * i + 1];
        float sv;
        if (pass) {
          sv = lae(l0, l1);
        } else {
          bool sm = sbm[cur * Nn + i] != 0, si = sbi[cur * Nn + i] != 0;
          float c0 = (sm || si)  ? NEGV : l0;
          float c1 = (sm || !si) ? NEGV : l1;
          sv = lae(c0, c1);
        }
        alpha[i * PITCH + j] = m + __logf(acc) + sv;
      }
    }
    __syncthreads();
  }

  if (tid == 0) {
    int len = 0;
    for (int j = 0; j < Nn; ++j) len += mspan[base + j]; // maskspan[b,0,:].sum()
    res[pass * Bn + b] = alpha[0 * PITCH + (len - 1)];
  }
}

__global__ void finalize_kernel(const unsigned char* __restrict__ mspan,
                                const float* __restrict__ res,
                                float* __restrict__ out) {
  if (threadIdx.x == 0 && blockIdx.x == 0) {
    float num = 0.0f; int den = 0;
    for (int b = 0; b < Bn; ++b) {
      const unsigned char* row = mspan + (size_t)b * Nn * Nn;
      int len = 0;
      for (int j = 0; j < Nn; ++j) len += row[j];
      den += len;
      num += res[Bn + b] - res[b]; // logz - marginals
    }
    out[0] = num / (float)den;
  }
}

extern "C" void kernel_launch(void* const* d_in, const int* in_sizes, int n_in,
                              void* d_out, int out_size, void* d_ws, size_t ws_size,
                              hipStream_t stream) {
  const float* logits         = (const float*)d_in[0];         // span_logits [16,256,256,2] f32
  const unsigned char* sind   = (const unsigned char*)d_in[1]; // spans_ind   bool
  const unsigned char* mspan  = (const unsigned char*)d_in[2]; // maskspan    bool
  const unsigned char* smask  = (const unsigned char*)d_in[3]; // span_mask   bool
  float* res = (float*)d_ws;                                   // 32 floats of scratch

  const size_t shbytes = (size_t)PITCH * Nn * 4   // chart: 263168
                       + (size_t)2 * Nn * 2 * 4   // staged logits: 4096
                       + (size_t)4 * Nn;          // staged masks: 1024  => 268288 <= 320KB WGP LDS
  dim3 grid(Bn, 2);
  cyk_kernel<<<grid, 512, shbytes, stream>>>(logits, sind, mspan, smask, res);
  finalize_kernel<<<1, 64, 0, stream>>>(mspan, res, (float*)d_out);
}